// CrossEntropyLoss_59777354826192
// MI455X (gfx1250) — compile-verified
//
#include <hip/hip_runtime.h>
#include <hip/hip_bf16.h>

#ifndef __has_builtin
#define __has_builtin(x) 0
#endif

#define CE_EPS 1e-8f

typedef int __attribute__((address_space(1))) as1_int;
typedef int __attribute__((address_space(3))) as3_int;

// ---- gfx1250 async wait (ASYNCcnt) --------------------------------------
__device__ __forceinline__ void wait_async0() {
#if __has_builtin(__builtin_amdgcn_s_wait_asynccnt)
    __builtin_amdgcn_s_wait_asynccnt(0);
#else
    asm volatile("s_wait_asynccnt 0" ::: "memory");
#endif
}

// ---- stage the 5 class weights into LDS via the CDNA5 async data path ----
__device__ __forceinline__ void stage_cw_to_lds(const float* __restrict__ cw,
                                                float* s_cw) {
#if __has_builtin(__builtin_amdgcn_global_load_async_to_lds_b32)
    if (threadIdx.x < 5) {
        __builtin_amdgcn_global_load_async_to_lds_b32(
            (as1_int*)(void*)const_cast<float*>(cw + threadIdx.x),
            (as3_int*)(void*)(s_cw + threadIdx.x),
            /*offset=*/0, /*cpol=*/0);
    }
#else
    if (threadIdx.x < 5) s_cw[threadIdx.x] = cw[threadIdx.x];
#endif
}

// ---------------------------------------------------------------------------
// Kernel 1: per-thread 4 pixels (float4 loads), block partial sums of
// weight_all * sum_c gold*log(pred+eps). Partial per block -> d_ws.
// ---------------------------------------------------------------------------
__global__ __launch_bounds__(256) void ce_partial_kernel(
    const float* __restrict__ pred,     // [C, HW]
    const float* __restrict__ gold,     // [C, HW]
    const float* __restrict__ weight,   // [HW]
    const float* __restrict__ cw,       // [C] (batch 0 of clss_weight_list)
    float* __restrict__ partial,        // [gridDim.x]
    int HW)
{
    __shared__ float s_cw[8];
    __shared__ float s_red[256];

    stage_cw_to_lds(cw, s_cw);
    wait_async0();          // data is in LDS once ASYNCcnt==0 in issuing wave
    __syncthreads();        // make visible to all waves in the workgroup

    const float w0 = s_cw[0];
    const float a1 = (s_cw[1] == 0.0f) ? w0 : s_cw[1];
    const float a2 = (s_cw[2] == 0.0f) ? w0 : s_cw[2];
    const float a3 = (s_cw[3] == 0.0f) ? w0 : s_cw[3];
    const float a4 = (s_cw[4] == 0.0f) ? w0 : s_cw[4];

    const int n4 = HW >> 2;                                  // float4 groups
    const int t4 = blockIdx.x * blockDim.x + threadIdx.x;    // this thread's group

    float acc = 0.0f;
    if (t4 < n4) {
        const float4* p4 = (const float4*)pred;
        const float4* g4 = (const float4*)gold;

        float pv[5][4], gv[5][4];
#pragma unroll
        for (int c = 0; c < 5; ++c) {
            float4 tp = p4[c * n4 + t4];
            float4 tg = g4[c * n4 + t4];
            pv[c][0] = tp.x; pv[c][1] = tp.y; pv[c][2] = tp.z; pv[c][3] = tp.w;
            gv[c][0] = tg.x; gv[c][1] = tg.y; gv[c][2] = tg.z; gv[c][3] = tg.w;
        }
        float4 wv4 = ((const float4*)weight)[t4];
        float wj[4] = {wv4.x, wv4.y, wv4.z, wv4.w};

#pragma unroll
        for (int j = 0; j < 4; ++j) {
            // argmax over channels (strict '>' keeps first occurrence,
            // matching jnp.argmax)
            float bp = pv[0][j]; int ap = 0;
            float bg = gv[0][j]; int ag = 0;
#pragma unroll
            for (int c = 1; c < 5; ++c) {
                if (pv[c][j] > bp) { bp = pv[c][j]; ap = c; }
                if (gv[c][j] > bg) { bg = gv[c][j]; ag = c; }
            }
            // soft CE term: sum_c gold * log(pred + eps)
            float ce = 0.0f;
#pragma unroll
            for (int c = 0; c < 5; ++c)
                ce += gv[c][j] * __logf(pv[c][j] + CE_EPS);

            // FP mask: true-class == background(0) and mispredicted
            const bool fp = (ag == 0) && (ap != 0);
            float wf = 0.0f;
            if (fp) wf = (ap == 1) ? a1 : (ap == 2) ? a2 : (ap == 3) ? a3 : a4;
            const float wa = (wf > 0.0f) ? wf : wj[j];

            acc += wa * ce;
        }
    }

    // deterministic block tree reduction
    s_red[threadIdx.x] = acc;
    __syncthreads();
#pragma unroll
    for (int s = 128; s > 0; s >>= 1) {
        if ((int)threadIdx.x < s) s_red[threadIdx.x] += s_red[threadIdx.x + s];
        __syncthreads();
    }
    if (threadIdx.x == 0) partial[blockIdx.x] = s_red[0];
}

// ---------------------------------------------------------------------------
// Kernel 2: reduce block partials, write mean loss = -sum/N.
// ---------------------------------------------------------------------------
__global__ __launch_bounds__(256) void ce_final_kernel(
    const float* __restrict__ partial, int n, float* __restrict__ out,
    float invN)
{
    __shared__ float s_red[256];
    float a = 0.0f;
    for (int i = threadIdx.x; i < n; i += 256) a += partial[i];
    s_red[threadIdx.x] = a;
    __syncthreads();
#pragma unroll
    for (int s = 128; s > 0; s >>= 1) {
        if ((int)threadIdx.x < s) s_red[threadIdx.x] += s_red[threadIdx.x + s];
        __syncthreads();
    }
    if (threadIdx.x == 0) out[0] = -s_red[0] * invN;
}

// ---------------------------------------------------------------------------
extern "C" void kernel_launch(void* const* d_in, const int* in_sizes, int n_in,
                              void* d_out, int out_size, void* d_ws, size_t ws_size,
                              hipStream_t stream) {
    const float* pred   = (const float*)d_in[0];  // [1,5,256,384]
    const float* gold   = (const float*)d_in[1];  // [1,5,256,384]
    const float* weight = (const float*)d_in[2];  // [1,256,384]
    const float* cw     = (const float*)d_in[3];  // [1,5]

    const int HW = in_sizes[2];          // B*H*W = 98304 (B==1)
    const int n4 = HW / 4;               // 24576 float4 groups
    const int threads = 256;
    const int blocks  = (n4 + threads - 1) / threads;   // 96

    float* partial = (float*)d_ws;

    ce_partial_kernel<<<blocks, threads, 0, stream>>>(
        pred, gold, weight, cw, partial, HW);
    ce_final_kernel<<<1, 256, 0, stream>>>(
        partial, blocks, (float*)d_out, 1.0f / (float)HW);
}